// uRNN_77352361001306
// MI455X (gfx1250) — compile-verified
//
#include <hip/hip_runtime.h>

// ---------------- CDNA5 types ----------------
typedef __attribute__((ext_vector_type(2))) float v2f;
typedef __attribute__((ext_vector_type(4))) float v4f;
typedef __attribute__((ext_vector_type(8))) float v8f;
typedef __attribute__((ext_vector_type(4))) int   v4i;

#define AS1 __attribute__((address_space(1)))
#define AS3 __attribute__((address_space(3)))

#if defined(__HIP_DEVICE_COMPILE__)
#if !__has_builtin(__builtin_amdgcn_wmma_f32_16x16x4_f32)
#error "missing __builtin_amdgcn_wmma_f32_16x16x4_f32 on this toolchain (device pass)"
#endif
#if __has_builtin(__builtin_amdgcn_global_load_async_to_lds_b128)
#define HAVE_ASYNC_LDS 1
#endif
#endif

// ---------------- problem constants ----------------
#define NN      1024
#define TT      512
#define DD      128
#define KTOT    (TT * DD)      // 65536
#define CHUNK_K 2048           // per-block K range (16 timesteps)
#define NCHUNK  32
#define NTILES  64             // 1024 / 16 rows
#define SUBK    256            // K per LDS stage
#define NSTAGE  (CHUNK_K / SUBK)
#define PADK    260            // +4 dword pad -> conflict-free b64 fragment reads

#define INV_SQRT_H 0.22360679775f          // 1/sqrt(20)
#define IF_SCALE   0.011180339887f         // 1/(20*sqrt(20))

// ws float layout
#define WS_COEF   0            // [512]
#define WS_U      512          // [128]
#define WS_BP     640          // [128]
#define WS_KONST  768          // [1]
#define WS_PART   1024         // [32*1024]

// ---------------- stage copy: async global->LDS when available ----------------
__device__ __forceinline__ void stage_copy(const float* gp, float* lp) {
#if defined(HAVE_ASYNC_LDS)
  // GLOBAL_LOAD_ASYNC_TO_LDS_B128: per-lane 16B, tracked by ASYNCcnt
  __builtin_amdgcn_global_load_async_to_lds_b128(
      (AS1 v4i*)gp, (AS3 v4i*)lp, 0, 0);
#else
  *(v4f*)lp = __builtin_nontemporal_load((const v4f*)gp);
#endif
}

__device__ __forceinline__ void stage_wait() {
#if defined(HAVE_ASYNC_LDS)
#if __has_builtin(__builtin_amdgcn_s_wait_asynccnt)
  __builtin_amdgcn_s_wait_asynccnt(0);
#else
  asm volatile("s_wait_asynccnt 0x0" ::: "memory");
#endif
#endif
}

// ---------------- setup: derive w, g, beta, u, b', coef, const ----------------
__global__ __launch_bounds__(256) void urnn_setup(
    const float* __restrict__ Cw,    // (40,128)
    const float* __restrict__ Bw,    // (40)
    const float* __restrict__ refl,  // (2,40)
    const float* __restrict__ theta, // (3,20)
    const float* __restrict__ mu0,   // (20)
    float* __restrict__ wsf)
{
  __shared__ float c0[20], s0[20], c2a[20], s2a[20], twc[20], tws[20];
  __shared__ float pr[20], pi[20];     // R0^T e  (= R0 e, R symmetric)
  __shared__ float q0r[20], q0i[20];   // D_theta1 applied to broadcast scalar 1
  __shared__ float fr[20], fi[20];     // iF(q0)
  __shared__ float wr_[20], wi_[20];   // w
  __shared__ float gr_[20], gi_[20];   // g
  __shared__ float sc[4];              // scale0, scale1, s1g, s2g

  const int t = threadIdx.x;
  if (t < 20) {
    float th0 = theta[t], th1 = theta[20 + t], th2 = theta[40 + t];
    c0[t] = cosf(th0);  s0[t] = sinf(th0);
    c2a[t] = cosf(th2); s2a[t] = sinf(th2);
    float c1 = cosf(th1), s1v = sinf(th1);
    q0r[t] = c1 - s1v;  q0i[t] = c1 + s1v;
    float ang = 0.31415926535897931f * (float)t;  // 2*pi*t/20
    twc[t] = cosf(ang); tws[t] = sinf(ang);
  }
  if (t == 0) {
    float v0 = 0.f, v1 = 0.f;
    for (int j = 0; j < 40; ++j) { v0 += refl[j] * refl[j]; v1 += refl[40 + j] * refl[40 + j]; }
    sc[0] = 2.f / v0; sc[1] = 2.f / v1;
  }
  __syncthreads();
  if (t < 20) {  // R0 applied to e_{20} (imag component 0)
    float s1 = -refl[20], s2 = refl[0], sca = sc[0];
    float vr = refl[t], vi = refl[20 + t];
    pr[t] = -sca * (s1 * vr + s2 * vi);
    pi[t] = (t == 0 ? 1.f : 0.f) - sca * (s2 * vr - s1 * vi);
  }
  __syncthreads();
  if (t < 20) {
    // F^T (= mult by conj DFT / sqrt(H)):  out_k = (1/sqrtH) sum_m c_m e^{+2pi i km/20}
    float ar = 0.f, ai = 0.f, br = 0.f, bi = 0.f;
    for (int m = 0; m < 20; ++m) {
      int idx = (t * m) % 20;
      float cc = twc[idx], ss = tws[idx];
      ar += pr[m] * cc - pi[m] * ss;   ai += pr[m] * ss + pi[m] * cc;
      br += q0r[m] * cc - q0i[m] * ss; bi += q0r[m] * ss + q0i[m] * cc;
    }
    ar *= INV_SQRT_H; ai *= INV_SQRT_H;
    // D0^T: rotate by -theta0
    wr_[t] = ar * c0[t] + ai * s0[t];
    wi_[t] = ai * c0[t] - ar * s0[t];
    // iF(q0): same +i kernel, scale 1/(20*sqrtH)
    fr[t] = br * IF_SCALE; fi[t] = bi * IF_SCALE;
  }
  __syncthreads();
  if (t == 0) {
    float s1 = 0.f, s2 = 0.f;
    for (int j = 0; j < 20; ++j) {
      s1 += fr[j] * refl[40 + j] - fi[j] * refl[60 + j];
      s2 += fr[j] * refl[60 + j] + fi[j] * refl[40 + j];
    }
    sc[2] = s1; sc[3] = s2;
  }
  __syncthreads();
  if (t < 20) {  // R1 then D_theta2  -> g
    float sca = sc[1], s1 = sc[2], s2 = sc[3];
    float vr = refl[40 + t], vi = refl[60 + t];
    float rr = fr[t] - sca * (s1 * vr + s2 * vi);
    float ri = fi[t] - sca * (s2 * vr - s1 * vi);
    gr_[t] = rr * c2a[t] - ri * s2a[t];
    gi_[t] = ri * c2a[t] + rr * s2a[t];
  }
  __syncthreads();
  if (t == 0) {
    float beta = 0.f, gB = 0.f, a0 = 0.f;
    for (int j = 0; j < 20; ++j) {
      beta += wr_[j] * gr_[j] + wi_[j] * gi_[j];
      gB   += gr_[j] * Bw[j] + gi_[j] * Bw[20 + j];
      a0   += mu0[j] * (wr_[j] - wi_[j]);   // w . (mu0, -mu0)
    }
    float cur = gB;                         // coef_t = gB * beta^(510-t)
    for (int tt = 510; tt >= 0; --tt) { wsf[WS_COEF + tt] = cur; cur *= beta; }
    wsf[WS_COEF + 511] = 0.f;
    wsf[WS_KONST] = cur * a0;               // gB * beta^511 * a0
  }
  __syncthreads();
  if (t < 128) {
    float uu = 0.f, bb = 0.f;
    for (int h = 0; h < 20; ++h)
      uu += wr_[h] * Cw[h * 128 + t] + wi_[h] * Cw[(20 + h) * 128 + t];
    for (int h = 0; h < 40; ++h)
      bb += Bw[h] * Cw[h * 128 + t];
    wsf[WS_U + t] = uu;
    wsf[WS_BP + t] = bb;
  }
}

// ---------------- main: streaming WMMA mat-vec, double-buffered LDS stages ----------------
__global__ __launch_bounds__(256) void urnn_main(
    const float* __restrict__ x,      // (1024, 512, 128)
    const float* __restrict__ wsf,
    float* __restrict__ partial)      // [32][1024]
{
  __shared__ float Vs[CHUNK_K];
  __shared__ float Xs[2][16 * PADK];
  __shared__ float red[128];
  __shared__ float vsum;

  const int tid = threadIdx.x;
  const int nt = blockIdx.x;          // 0..63  (16 samples each)
  const int kc = blockIdx.y;          // 0..31  (16 timesteps each)
  const float* coef = wsf + WS_COEF;
  const float* u    = wsf + WS_U;
  const float* bp   = wsf + WS_BP;

  // Build V chunk: V[t,d] = coef[t]*u[d] + (t==511)*bp[d]
  float loc = 0.f;
#pragma unroll
  for (int j = 0; j < 8; ++j) {
    int k = tid + j * 256;
    int tg = kc * 16 + (k >> 7);
    int d  = k & 127;
    float v = coef[tg] * u[d];
    if (tg == 511) v += bp[d];
    Vs[k] = v;
    loc += fabsf(v);
  }
  if (tid == 0) vsum = 0.f;
  __syncthreads();
  atomicAdd(&vsum, loc);
  __syncthreads();
  if (vsum == 0.f) {                  // beta^k underflowed: contribution exactly zero
    if (tid < 16) partial[kc * 1024 + nt * 16 + tid] = 0.f;
    return;
  }

  const int lane = tid & 31, wave = tid >> 5;
  const int row = lane & 15, hi = lane >> 4;
  const int lrow = tid >> 4, j16 = tid & 15;
  const size_t gbase = (size_t)(nt * 16 + lrow) * (size_t)KTOT + (size_t)kc * CHUNK_K;

  v8f acc = {0.f, 0.f, 0.f, 0.f, 0.f, 0.f, 0.f, 0.f};

  // prologue: stage 0 -> buffer 0
#pragma unroll
  for (int it = 0; it < 4; ++it) {
    int col = j16 * 4 + it * 64;
    stage_copy(x + gbase + (size_t)col, &Xs[0][lrow * PADK + col]);
  }

  for (int s = 0; s < NSTAGE; ++s) {
    stage_wait();                     // own async fills complete
    __syncthreads();                  // all fills visible; prev-stage readers done
    if (s + 1 < NSTAGE) {             // prefetch next stage into the other buffer
      const int bn = (s + 1) & 1;
#pragma unroll
      for (int it = 0; it < 4; ++it) {
        int col = j16 * 4 + it * 64;
        stage_copy(x + gbase + (size_t)((s + 1) * SUBK + col),
                   &Xs[bn][lrow * PADK + col]);
      }
    }
    const int buf = s & 1;
    // each wave: 32 K-slots -> 8 x V_WMMA_F32_16X16X4_F32
#pragma unroll
    for (int m = 0; m < 8; ++m) {
      int kk = wave * 32 + m * 4;
      v2f a = *(const v2f*)&Xs[buf][row * PADK + kk + 2 * hi]; // A[16x4] lane layout
      int kv = s * SUBK + kk;
      v2f b;
      b.x = Vs[kv + hi];        // B row K=0/K=1 across lane halves
      b.y = Vs[kv + 2 + hi];    // B row K=2/K=3
#if defined(__HIP_DEVICE_COMPILE__)
      acc = __builtin_amdgcn_wmma_f32_16x16x4_f32(false, a, false, b,
                                                  (short)0, acc, false, false);
#else
      (void)a; (void)b;   // host pass: never executed
#endif
    }
  }
  __syncthreads();
  // column 0 of C: lane0 -> rows 0..7 (v0..v7), lane16 -> rows 8..15
  if (lane == 0 || lane == 16) {
#pragma unroll
    for (int r = 0; r < 8; ++r) red[wave * 16 + hi * 8 + r] = acc[r];
  }
  __syncthreads();
  if (tid < 16) {
    float ssum = 0.f;
    for (int w = 0; w < 8; ++w) ssum += red[w * 16 + tid];
    partial[kc * 1024 + nt * 16 + tid] = ssum;
  }
}

// ---------------- reduce: y[n] = const + sum_kc partial ----------------
__global__ __launch_bounds__(256) void urnn_reduce(const float* __restrict__ wsf,
                                                   float* __restrict__ y)
{
  int n = blockIdx.x * 256 + threadIdx.x;
  float s = wsf[WS_KONST];
  const float* partial = wsf + WS_PART;
  for (int kc = 0; kc < NCHUNK; ++kc) s += partial[kc * 1024 + n];
  y[n] = s;
}

// ---------------- launch ----------------
extern "C" void kernel_launch(void* const* d_in, const int* in_sizes, int n_in,
                              void* d_out, int out_size, void* d_ws, size_t ws_size,
                              hipStream_t stream) {
  const float* x     = (const float*)d_in[0];  // inputs (1024,512,128)
  const float* Cw    = (const float*)d_in[1];  // (40,128)
  const float* Bw    = (const float*)d_in[2];  // (1,40)
  const float* refl  = (const float*)d_in[3];  // (2,40)
  const float* theta = (const float*)d_in[4];  // (3,20)
  const float* mu0   = (const float*)d_in[5];  // (20)
  float* wsf = (float*)d_ws;                   // needs ~132 KB

  urnn_setup<<<1, 256, 0, stream>>>(Cw, Bw, refl, theta, mu0, wsf);
  urnn_main<<<dim3(NTILES, NCHUNK), 256, 0, stream>>>(x, wsf, wsf + WS_PART);
  urnn_reduce<<<4, 256, 0, stream>>>(wsf, (float*)d_out);
}